// FairGNN_3_38740605010064
// MI455X (gfx1250) — compile-verified
//
#include <hip/hip_runtime.h>
#include <cfloat>

#define N_NODES 100000
#define NFEAT   256
#define NHID    64
#define NEDGE   1600000
#define TOTE    (NEDGE + N_NODES)
#define MTILES  (N_NODES / 16)          // 6250, exact
#define EPSBN   1e-5f
#define NEG_SLOPE 0.2f

typedef __attribute__((ext_vector_type(2))) float v2f;
typedef __attribute__((ext_vector_type(4))) float v4f;
typedef __attribute__((ext_vector_type(8))) float v8f;

// monotone float<->uint mapping so float max == unsigned max
__device__ __forceinline__ unsigned ford(float f) {
    unsigned u = __float_as_uint(f);
    return (u & 0x80000000u) ? ~u : (u | 0x80000000u);
}
__device__ __forceinline__ float dorf(unsigned u) {
    return (u & 0x80000000u) ? __uint_as_float(u & 0x7fffffffu) : __uint_as_float(~u);
}
__device__ __forceinline__ float leaky(float v) {
    return fmaxf(v, 0.f) + NEG_SLOPE * fminf(v, 0.f);
}

// ---------------- init: zero accumulators, set segment-max to -inf ----------
__global__ void k_init(float* acc_est, float* acc_gnn,
                       unsigned* m_e, unsigned* m_g, float* z_e, float* z_g) {
    long i = (long)blockIdx.x * blockDim.x + threadIdx.x;
    long tot = (long)N_NODES * NHID;
    if (i < tot) { acc_est[i] = 0.f; acc_gnn[i] = 0.f; }
    if (i < N_NODES) {
        m_e[i] = ford(-FLT_MAX);
        m_g[i] = ford(-FLT_MAX);
        z_e[i] = 0.f;
        z_g[i] = 0.f;
    }
}

// ---------------- WMMA GEMM: [N,256] x ([256,64]|[256,64]) -> h_est,h_gnn ---
// wave owns 16 rows of g and all 8 N-tiles (4 est + 4 gnn): g read once.
// f32 WMMA 16x16x4: A 16x4 (v2f/lane), B 4x16 (v2f/lane), C 16x16 (v8f/lane).
__global__ __launch_bounds__(256) void k_gemm(const float* __restrict__ g,
                                              const float* __restrict__ We,
                                              const float* __restrict__ Wg,
                                              float* __restrict__ h_est,
                                              float* __restrict__ h_gnn) {
    int wave = threadIdx.x >> 5;
    int lane = threadIdx.x & 31;
    int mtile = blockIdx.x * 8 + wave;
    if (mtile >= MTILES) return;            // whole-wave guard (EXEC all 1s for WMMA)
    int m0   = mtile * 16;
    int lrow = lane & 15;                   // M (A) / N (B,C) within tile
    int lhi  = lane >> 4;                   // K-half select

    const float* grow = g + (long)(m0 + lrow) * NFEAT;

    v8f acc[8] = {};
    for (int k = 0; k < NFEAT; k += 4) {
        // A: lanes 0-15 hold K=k..k+1, lanes 16-31 hold K=k+2..k+3 (ISA 16x4 layout)
        v2f a = *(const v2f*)(grow + k + 2 * lhi);
#pragma unroll
        for (int t = 0; t < 8; ++t) {
            const float* Wp = (t < 4) ? We : Wg;
            int c  = (t & 3) * 16 + lrow;
            int kr = k + 2 * lhi;
            v2f b;
            b.x = Wp[(long)kr * NHID + c];
            b.y = Wp[(long)(kr + 1) * NHID + c];
            acc[t] = __builtin_amdgcn_wmma_f32_16x16x4_f32(
                false, a, false, b, (short)0, acc[t], false, false);
        }
    }
#pragma unroll
    for (int t = 0; t < 8; ++t) {
        float* dst = (t < 4) ? h_est : h_gnn;
        int c = (t & 3) * 16 + lrow;
#pragma unroll
        for (int r = 0; r < 8; ++r) {       // C: vgpr r -> row m0+r (+8 for hi lanes)
            dst[(long)(m0 + r + 8 * lhi) * NHID + c] = acc[t][r];
        }
    }
}

// ---------------- per-node attention logits: 4 length-64 dots --------------
__global__ void k_eprep(const float* __restrict__ h_est, const float* __restrict__ h_gnn,
                        const float* __restrict__ ase, const float* __restrict__ ade,
                        const float* __restrict__ asg, const float* __restrict__ adg,
                        float* es_e, float* ed_e, float* es_g, float* ed_g) {
    int n = blockIdx.x * blockDim.x + threadIdx.x;
    if (n >= N_NODES) return;
    const float* he = h_est + (long)n * NHID;
    const float* hg = h_gnn + (long)n * NHID;
    float se = 0.f, de = 0.f, sg = 0.f, dg = 0.f;
#pragma unroll 4
    for (int f = 0; f < NHID; f += 4) {
        v4f e  = *(const v4f*)(he + f);
        v4f g4 = *(const v4f*)(hg + f);
        v4f a1 = *(const v4f*)(ase + f);
        v4f a2 = *(const v4f*)(ade + f);
        v4f a3 = *(const v4f*)(asg + f);
        v4f a4 = *(const v4f*)(adg + f);
        se += e.x * a1.x + e.y * a1.y + e.z * a1.z + e.w * a1.w;
        de += e.x * a2.x + e.y * a2.y + e.z * a2.z + e.w * a2.w;
        sg += g4.x * a3.x + g4.y * a3.y + g4.z * a3.z + g4.w * a3.w;
        dg += g4.x * a4.x + g4.y * a4.y + g4.z * a4.z + g4.w * a4.w;
    }
    es_e[n] = se; ed_e[n] = de; es_g[n] = sg; ed_g[n] = dg;
}

// ---------------- pass 1: segment max over dst (uint-ordered atomicMax) ----
__global__ void k_max(const int* __restrict__ x,
                      const float* __restrict__ es_e, const float* __restrict__ ed_e,
                      const float* __restrict__ es_g, const float* __restrict__ ed_g,
                      unsigned* m_e, unsigned* m_g) {
    long e = (long)blockIdx.x * blockDim.x + threadIdx.x;
    if (e >= TOTE) return;
    int s, d;
    if (e < NEDGE) { s = x[e]; d = x[NEDGE + e]; } else { s = d = (int)(e - NEDGE); }
    atomicMax(m_e + d, ford(leaky(es_e[s] + ed_e[d])));
    atomicMax(m_g + d, ford(leaky(es_g[s] + ed_g[d])));
}

// ---------------- pass 2: segment sum of exp(v - max) ----------------------
__global__ void k_sum(const int* __restrict__ x,
                      const float* __restrict__ es_e, const float* __restrict__ ed_e,
                      const float* __restrict__ es_g, const float* __restrict__ ed_g,
                      const unsigned* __restrict__ m_e, const unsigned* __restrict__ m_g,
                      float* z_e, float* z_g) {
    long e = (long)blockIdx.x * blockDim.x + threadIdx.x;
    if (e >= TOTE) return;
    int s, d;
    if (e < NEDGE) { s = x[e]; d = x[NEDGE + e]; } else { s = d = (int)(e - NEDGE); }
    atomicAdd(z_e + d, __expf(leaky(es_e[s] + ed_e[d]) - dorf(m_e[d])));
    atomicAdd(z_g + d, __expf(leaky(es_g[s] + ed_g[d]) - dorf(m_g[d])));
}

// ---------------- pass 3: weighted scatter, one wave per edge --------------
// lane l handles features 2l,2l+1 for both GATs; h + accumulators are L2-resident.
__global__ __launch_bounds__(256) void k_scatter(const int* __restrict__ x,
        const float* __restrict__ es_e, const float* __restrict__ ed_e,
        const float* __restrict__ es_g, const float* __restrict__ ed_g,
        const unsigned* __restrict__ m_e, const unsigned* __restrict__ m_g,
        const float* __restrict__ z_e, const float* __restrict__ z_g,
        const float* __restrict__ h_est, const float* __restrict__ h_gnn,
        float* acc_est, float* acc_gnn) {
    int wave = threadIdx.x >> 5;
    int lane = threadIdx.x & 31;
    long e = (long)blockIdx.x * 8 + wave;
    if (e >= TOTE) return;
    int s, d;
    if (e < NEDGE) { s = x[e]; d = x[NEDGE + e]; } else { s = d = (int)(e - NEDGE); }
    float ae = __expf(leaky(es_e[s] + ed_e[d]) - dorf(m_e[d])) / z_e[d];
    float ag = __expf(leaky(es_g[s] + ed_g[d]) - dorf(m_g[d])) / z_g[d];
    int f0 = lane * 2;
    v2f he = *(const v2f*)(h_est + (long)s * NHID + f0);
    v2f hg = *(const v2f*)(h_gnn + (long)s * NHID + f0);
    float* pe = acc_est + (long)d * NHID + f0;
    float* pg = acc_gnn + (long)d * NHID + f0;
    atomicAdd(pe,     he.x * ae);
    atomicAdd(pe + 1, he.y * ae);
    atomicAdd(pg,     hg.x * ag);
    atomicAdd(pg + 1, hg.y * ag);
}

// ---------------- finalize: bias->ReLU->BN (both) + classifier -------------
__global__ void k_final(float* d_out, const float* __restrict__ acc_gnn,
        const float* __restrict__ eb, const float* __restrict__ egam,
        const float* __restrict__ ebeta, const float* __restrict__ erm,
        const float* __restrict__ erv,
        const float* __restrict__ gb, const float* __restrict__ ggam,
        const float* __restrict__ gbeta, const float* __restrict__ grm,
        const float* __restrict__ grv,
        const float* __restrict__ clsW, const float* __restrict__ clsb) {
    int n = blockIdx.x * blockDim.x + threadIdx.x;
    if (n >= N_NODES) return;
    float* srow = d_out + N_NODES + (long)n * NHID;   // est accumulators in-place
    const float* grow = acc_gnn + (long)n * NHID;
    float y = 0.f;
#pragma unroll 4
    for (int f = 0; f < NHID; ++f) {
        float a = fmaxf(srow[f] + eb[f], 0.f);
        srow[f] = (a - erm[f]) * rsqrtf(erv[f] + EPSBN) * egam[f] + ebeta[f];
        float b = fmaxf(grow[f] + gb[f], 0.f);
        float zv = (b - grm[f]) * rsqrtf(grv[f] + EPSBN) * ggam[f] + gbeta[f];
        y += zv * clsW[f];
    }
    d_out[n] = y + clsb[0];
}

extern "C" void kernel_launch(void* const* d_in, const int* in_sizes, int n_in,
                              void* d_out, int out_size, void* d_ws, size_t ws_size,
                              hipStream_t stream) {
    const float* g        = (const float*)d_in[0];
    const int*   x        = (const int*)d_in[1];
    const float* est_W    = (const float*)d_in[2];
    const float* est_asrc = (const float*)d_in[3];
    const float* est_adst = (const float*)d_in[4];
    const float* est_bias = (const float*)d_in[5];
    const float* est_gam  = (const float*)d_in[6];
    const float* est_beta = (const float*)d_in[7];
    const float* est_rm   = (const float*)d_in[8];
    const float* est_rv   = (const float*)d_in[9];
    const float* gnn_W    = (const float*)d_in[10];
    const float* gnn_asrc = (const float*)d_in[11];
    const float* gnn_adst = (const float*)d_in[12];
    const float* gnn_bias = (const float*)d_in[13];
    const float* gnn_gam  = (const float*)d_in[14];
    const float* gnn_beta = (const float*)d_in[15];
    const float* gnn_rm   = (const float*)d_in[16];
    const float* gnn_rv   = (const float*)d_in[17];
    const float* cls_W    = (const float*)d_in[18];
    const float* cls_b    = (const float*)d_in[19];
    float* out = (float*)d_out;                // [0,N): y ; [N, N+N*64): s

    // workspace layout (floats): 200*N total = 80 MB
    float*    ws     = (float*)d_ws;
    float*    h_est  = ws;                       // N*64
    float*    h_gnn  = h_est + (long)N_NODES * NHID;
    float*    acc_g  = h_gnn + (long)N_NODES * NHID;
    float*    es_e   = acc_g + (long)N_NODES * NHID;
    float*    ed_e   = es_e + N_NODES;
    float*    es_g   = ed_e + N_NODES;
    float*    ed_g   = es_g + N_NODES;
    unsigned* m_e    = (unsigned*)(ed_g + N_NODES);
    unsigned* m_g    = m_e + N_NODES;
    float*    z_e    = (float*)(m_g + N_NODES);
    float*    z_g    = z_e + N_NODES;
    float*    acc_e  = out + N_NODES;            // est scatter accumulates in d_out

    const int BT = 256;
    long initN = (long)N_NODES * NHID;
    k_init<<<(unsigned)((initN + BT - 1) / BT), BT, 0, stream>>>(acc_e, acc_g, m_e, m_g, z_e, z_g);

    k_gemm<<<(MTILES + 7) / 8, BT, 0, stream>>>(g, est_W, gnn_W, h_est, h_gnn);

    k_eprep<<<(N_NODES + BT - 1) / BT, BT, 0, stream>>>(h_est, h_gnn,
            est_asrc, est_adst, gnn_asrc, gnn_adst, es_e, ed_e, es_g, ed_g);

    k_max<<<(TOTE + BT - 1) / BT, BT, 0, stream>>>(x, es_e, ed_e, es_g, ed_g, m_e, m_g);

    k_sum<<<(TOTE + BT - 1) / BT, BT, 0, stream>>>(x, es_e, ed_e, es_g, ed_g, m_e, m_g, z_e, z_g);

    k_scatter<<<(TOTE + 7) / 8, BT, 0, stream>>>(x, es_e, ed_e, es_g, ed_g,
            m_e, m_g, z_e, z_g, h_est, h_gnn, acc_e, acc_g);

    k_final<<<(N_NODES + BT - 1) / BT, BT, 0, stream>>>(out, acc_g,
            est_bias, est_gam, est_beta, est_rm, est_rv,
            gnn_bias, gnn_gam, gnn_beta, gnn_rm, gnn_rv, cls_W, cls_b);
}